// MambaRec_84499186581567
// MI455X (gfx1250) — compile-verified
//
#include <hip/hip_runtime.h>
#include <hip/hip_bf16.h>
#include <math.h>

// ---------------- shapes ----------------
#define BATCH     32
#define MAXLEN    128
#define D_MODEL   64
#define D_STATE   32
#define D_CONV    4
#define D_INNER   128
#define DT_RANK   4
#define NTOK      (BATCH * MAXLEN)        // 4096
#define VOCAB     50001

typedef __attribute__((ext_vector_type(2))) float v2f;
typedef __attribute__((ext_vector_type(8))) float v8f;

__device__ __forceinline__ float silu_f(float v) {
    return v / (1.0f + __expf(-v));
}
__device__ __forceinline__ float softplus_f(float v) {
    return (v > 20.0f) ? v : log1pf(__expf(v));
}

// ------------------------------------------------------------------
// K1: embedding gather + positional + pad mask  ->  x (4096 x 64)
// ------------------------------------------------------------------
__global__ void k_embed(const int* __restrict__ ids,
                        const float* __restrict__ item_emb,
                        const float* __restrict__ pos_emb,
                        float* __restrict__ x) {
    int idx = blockIdx.x * 256 + threadIdx.x;        // 262144 total
    int m = idx >> 6;
    int d = idx & 63;
    int l = m & (MAXLEN - 1);
    int tok = ids[m];
    float v = 0.0f;
    if (tok != 0)
        v = item_emb[(size_t)tok * D_MODEL + d] * 8.0f + pos_emb[l * D_MODEL + d];
    x[idx] = v;
}

// ------------------------------------------------------------------
// K2: in_proj  xz = x @ in_proj_w^T  ->  xcpre (4096x128), z (4096x128)
// one block per token row, 256 threads (one per output feature)
// ------------------------------------------------------------------
__global__ void k_inproj(const float* __restrict__ x,
                         const float* __restrict__ W,      // (256,64)
                         float* __restrict__ xcpre,
                         float* __restrict__ zbuf) {
    __shared__ float xs[D_MODEL];
    int m = blockIdx.x;
    int e = threadIdx.x;
    if (e < D_MODEL) xs[e] = x[m * D_MODEL + e];
    __syncthreads();
    const float* w = W + e * D_MODEL;
    float acc = 0.0f;
#pragma unroll
    for (int d = 0; d < D_MODEL; ++d) acc = fmaf(xs[d], w[d], acc);
    if (e < D_INNER) xcpre[m * D_INNER + e] = acc;
    else             zbuf[m * D_INNER + (e - D_INNER)] = acc;
}

// ------------------------------------------------------------------
// K3: causal depthwise conv1d (k=4) + bias + SiLU -> xc (4096x128)
// ------------------------------------------------------------------
__global__ void k_conv(const float* __restrict__ xcpre,
                       const float* __restrict__ cw,       // (128,4)
                       const float* __restrict__ cb,
                       float* __restrict__ xc) {
    int idx = blockIdx.x * 256 + threadIdx.x;        // 524288 total
    int m = idx >> 7;
    int c = idx & (D_INNER - 1);
    int l = m & (MAXLEN - 1);
    float acc = cb[c];
#pragma unroll
    for (int k = 0; k < D_CONV; ++k) {
        int ll = l + k - (D_CONV - 1);
        float v = (ll >= 0) ? xcpre[(m + k - (D_CONV - 1)) * D_INNER + c] : 0.0f;
        acc = fmaf(v, cw[c * D_CONV + k], acc);
    }
    xc[idx] = silu_f(acc);
}

// ------------------------------------------------------------------
// K4: x_proj (-> dt,B,C) and delta = softplus(dt @ dt_proj_w^T + b)
// one block per token row, 128 threads
// ------------------------------------------------------------------
__global__ void k_xproj(const float* __restrict__ xc,
                        const float* __restrict__ xpw,     // (68,128)
                        const float* __restrict__ dtw,     // (128,4)
                        const float* __restrict__ dtb,     // (128,)
                        float* __restrict__ Bm,            // (4096,32)
                        float* __restrict__ Cm,            // (4096,32)
                        float* __restrict__ delta) {       // (4096,128)
    __shared__ float xs[D_INNER];
    __shared__ float dts[DT_RANK];
    int m = blockIdx.x;
    int t = threadIdx.x;
    xs[t] = xc[m * D_INNER + t];
    __syncthreads();
    if (t < DT_RANK + 2 * D_STATE) {
        const float* w = xpw + t * D_INNER;
        float acc = 0.0f;
#pragma unroll 8
        for (int c = 0; c < D_INNER; ++c) acc = fmaf(xs[c], w[c], acc);
        if (t < DT_RANK)                dts[t] = acc;
        else if (t < DT_RANK + D_STATE) Bm[m * D_STATE + (t - DT_RANK)] = acc;
        else                            Cm[m * D_STATE + (t - DT_RANK - D_STATE)] = acc;
    }
    __syncthreads();
    float acc = dtb[t];
#pragma unroll
    for (int r = 0; r < DT_RANK; ++r) acc = fmaf(dts[r], dtw[t * DT_RANK + r], acc);
    delta[m * D_INNER + t] = softplus_f(acc);
}

// ------------------------------------------------------------------
// K5: selective scan. one block per batch, 128 threads (one per channel),
// 32-state recurrence kept in VGPRs; B/C broadcast through LDS.
// ------------------------------------------------------------------
__global__ void k_scan(const float* __restrict__ xc,
                       const float* __restrict__ zbuf,
                       const float* __restrict__ delta,
                       const float* __restrict__ Bm,
                       const float* __restrict__ Cm,
                       const float* __restrict__ A_log,    // (128,32)
                       const float* __restrict__ Dp,       // (128,)
                       float* __restrict__ y) {            // (4096,128)
    __shared__ float Bs[D_STATE];
    __shared__ float Cs[D_STATE];
    int b = blockIdx.x;
    int c = threadIdx.x;
    float a[D_STATE], h[D_STATE];
#pragma unroll
    for (int s = 0; s < D_STATE; ++s) {
        a[s] = -__expf(A_log[c * D_STATE + s]);
        h[s] = 0.0f;
    }
    float dp = Dp[c];
    for (int t = 0; t < MAXLEN; ++t) {
        int m = b * MAXLEN + t;
        if (c < D_STATE) {
            Bs[c] = Bm[m * D_STATE + c];
            Cs[c] = Cm[m * D_STATE + c];
        }
        __syncthreads();
        float dlt = delta[m * D_INNER + c];
        float xv  = xc[m * D_INNER + c];
        float dx  = dlt * xv;
        float acc = 0.0f;
#pragma unroll
        for (int s = 0; s < D_STATE; ++s) {
            h[s] = fmaf(__expf(dlt * a[s]), h[s], dx * Bs[s]);
            acc  = fmaf(h[s], Cs[s], acc);
        }
        float zz = zbuf[m * D_INNER + c];
        float yv = fmaf(dp, xv, acc) * silu_f(zz);
        y[m * D_INNER + c] = yv;
        __syncthreads();
    }
}

// ------------------------------------------------------------------
// K6: out_proj  seqs2 = y @ out_proj_w^T   (4096x128 . 128x64)
// one block per token row, 64 threads
// ------------------------------------------------------------------
__global__ void k_outproj(const float* __restrict__ y,
                          const float* __restrict__ W,     // (64,128)
                          float* __restrict__ s2) {        // (4096,64)
    __shared__ float ys[D_INNER];
    int m = blockIdx.x;
    int t = threadIdx.x;
    ys[t]      = y[m * D_INNER + t];
    ys[t + 64] = y[m * D_INNER + t + 64];
    __syncthreads();
    const float* w = W + t * D_INNER;
    float acc = 0.0f;
#pragma unroll 8
    for (int c = 0; c < D_INNER; ++c) acc = fmaf(ys[c], w[c], acc);
    s2[m * D_MODEL + t] = acc;
}

// ------------------------------------------------------------------
// K7: logits = s2 @ item_emb^T   via V_WMMA_F32_16X16X4_F32
// grid: one block per 32-wide vocab stripe (1563 blocks), 256 thr = 8 waves.
// Each wave owns TWO 16-col vocab tiles: B fragments loaded once
// (2 x 16 x v2f = 64 VGPRs), then sweeps 32 m-tiles. Per m-tile:
// 16 b64 A-fragment loads feed 32 WMMAs split across two independent
// accumulator chains (c0/c1) -> 2x ILP on the matrix pipe, and A-tile
// L2 traffic is halved vs one-tile-per-block.
// A frag layout (ISA 7.12.2, 32-bit A 16x4): lane%16 = M,
//   vgpr pair = {k, k+1} with k = kb*4 + 2*(lane/16).
// C/D layout: vgpr i -> M = m0 + 8*(lane/16) + i, N = n0 + lane%16.
// ------------------------------------------------------------------
__global__ __launch_bounds__(256)
void k_logits_wmma(const float* __restrict__ S,     // (4096,64)
                   const float* __restrict__ E,     // (50001,64)
                   float* __restrict__ out) {       // (4096,50001)
    const int V = VOCAB;
    int lane = threadIdx.x & 31;
    int wave = threadIdx.x >> 5;
    int nl   = lane & 15;
    int hi   = lane >> 4;
    int n0   = blockIdx.x * 32;
    int na   = n0 + nl;          // tile 0 column
    int nb   = n0 + 16 + nl;     // tile 1 column

    // B fragments: B[k][n] = E[n][k]; clamp OOB rows (stores are guarded).
    const float* erow0 = E + (size_t)((na < V) ? na : (V - 1)) * D_MODEL;
    const float* erow1 = E + (size_t)((nb < V) ? nb : (V - 1)) * D_MODEL;
    v2f bf0[16], bf1[16];
#pragma unroll
    for (int kb = 0; kb < 16; ++kb) {
        int k = kb * 4 + hi * 2;
        bf0[kb].x = erow0[k];
        bf0[kb].y = erow0[k + 1];
        bf1[kb].x = erow1[k];
        bf1[kb].y = erow1[k + 1];
    }

    for (int mt = wave; mt < NTOK / 16; mt += 8) {
        int m0 = mt * 16;
        const float* srow = S + (size_t)(m0 + nl) * D_MODEL;
        v2f af[16];
#pragma unroll
        for (int kb = 0; kb < 16; ++kb) {
            int k = kb * 4 + hi * 2;
            af[kb].x = srow[k];
            af[kb].y = srow[k + 1];
        }
        v8f c0 = {};
        v8f c1 = {};
#pragma unroll
        for (int kb = 0; kb < 16; ++kb) {
            c0 = __builtin_amdgcn_wmma_f32_16x16x4_f32(
                     false, af[kb], false, bf0[kb], (short)0, c0, false, false);
            c1 = __builtin_amdgcn_wmma_f32_16x16x4_f32(
                     false, af[kb], false, bf1[kb], (short)0, c1, false, false);
        }
        size_t mrow = (size_t)(m0 + hi * 8) * V;
        if (na < V) {
            size_t base = mrow + na;
            out[base + 0 * (size_t)V] = c0[0];
            out[base + 1 * (size_t)V] = c0[1];
            out[base + 2 * (size_t)V] = c0[2];
            out[base + 3 * (size_t)V] = c0[3];
            out[base + 4 * (size_t)V] = c0[4];
            out[base + 5 * (size_t)V] = c0[5];
            out[base + 6 * (size_t)V] = c0[6];
            out[base + 7 * (size_t)V] = c0[7];
        }
        if (nb < V) {
            size_t base = mrow + nb;
            out[base + 0 * (size_t)V] = c1[0];
            out[base + 1 * (size_t)V] = c1[1];
            out[base + 2 * (size_t)V] = c1[2];
            out[base + 3 * (size_t)V] = c1[3];
            out[base + 4 * (size_t)V] = c1[4];
            out[base + 5 * (size_t)V] = c1[5];
            out[base + 6 * (size_t)V] = c1[6];
            out[base + 7 * (size_t)V] = c1[7];
        }
    }
}

// ------------------------------------------------------------------
extern "C" void kernel_launch(void* const* d_in, const int* in_sizes, int n_in,
                              void* d_out, int out_size, void* d_ws, size_t ws_size,
                              hipStream_t stream) {
    const int*   ids       = (const int*)  d_in[0];
    const float* item_emb  = (const float*)d_in[1];
    const float* pos_emb   = (const float*)d_in[2];
    const float* in_proj_w = (const float*)d_in[3];
    const float* conv_w    = (const float*)d_in[4];
    const float* conv_b    = (const float*)d_in[5];
    const float* x_proj_w  = (const float*)d_in[6];
    const float* dt_proj_w = (const float*)d_in[7];
    const float* dt_proj_b = (const float*)d_in[8];
    const float* A_log     = (const float*)d_in[9];
    const float* D_param   = (const float*)d_in[10];
    const float* out_proj_w= (const float*)d_in[11];
    float* out = (float*)d_out;

    // workspace carve-up (floats)
    float* ws     = (float*)d_ws;
    float* x      = ws;                                   // 4096*64
    float* xcpre  = x      + NTOK * D_MODEL;              // 4096*128
    float* zbuf   = xcpre  + NTOK * D_INNER;              // 4096*128
    float* xc     = zbuf   + NTOK * D_INNER;              // 4096*128
    float* Bm     = xc     + NTOK * D_INNER;              // 4096*32
    float* Cm     = Bm     + NTOK * D_STATE;              // 4096*32
    float* delta  = Cm     + NTOK * D_STATE;              // 4096*128
    float* y      = delta  + NTOK * D_INNER;              // 4096*128
    float* s2     = y      + NTOK * D_INNER;              // 4096*64

    k_embed  <<<(NTOK * D_MODEL) / 256, 256, 0, stream>>>(ids, item_emb, pos_emb, x);
    k_inproj <<<NTOK, 256, 0, stream>>>(x, in_proj_w, xcpre, zbuf);
    k_conv   <<<(NTOK * D_INNER) / 256, 256, 0, stream>>>(xcpre, conv_w, conv_b, xc);
    k_xproj  <<<NTOK, 128, 0, stream>>>(xc, x_proj_w, dt_proj_w, dt_proj_b, Bm, Cm, delta);
    k_scan   <<<BATCH, 128, 0, stream>>>(xc, zbuf, delta, Bm, Cm, A_log, D_param, y);
    k_outproj<<<NTOK, 64, 0, stream>>>(y, out_proj_w, s2);
    k_logits_wmma<<<(VOCAB + 31) / 32, 256, 0, stream>>>(s2, item_emb, out);
}